// MagneticAdaptiveplusGraphConvolution_83391085019492
// MI455X (gfx1250) — compile-verified
//
#include <hip/hip_runtime.h>
#include <math.h>

// ---------------------------------------------------------------------------
// MagneticAdaptive+ Graph Convolution for MI455X (gfx1250).
//
// N=50000, E=800000, F=H=128, O=16, 3 propagation steps, 2 MLP layers.
//
// Phase plan:
//   1) edge_weights_kernel    : w_r/w_i per edge (E threads)
//   2) init_slice0_kernel     : copy input features into step-0 slices
//      + hipMemsetAsync       : zero accumulation slices 1..3
//   3) prop_step_kernel x3    : fused complex SpMM (1 wave per edge,
//                               float4 gather + 8 fp32 atomics/lane; all
//                               feature matrices are L2-resident: 102MB<192MB)
//   4) attn_pool_kernel x2    : wave-per-node dot + 4-way softmax (wave32
//                               __shfl_xor reductions)
//   5) mlp_layer_kernel x2    : fused real+imag fp32 GEMM via
//                               V_WMMA_F32_16X16X4_F32 with in-register
//                               ComReLU masking; A staged in LDS
//   6) out_layer_kernel       : [Xr|Xi] @ out_w (K=256) via f32 WMMA
// ---------------------------------------------------------------------------

typedef float v2f __attribute__((ext_vector_type(2)));
typedef float v8f __attribute__((ext_vector_type(8)));

static __device__ __forceinline__ v8f wmma_f32_k4(v2f a, v2f b, v8f c) {
  // D = A(16x4) * B(4x16) + C(16x16), full fp32
  return __builtin_amdgcn_wmma_f32_16x16x4_f32(
      /*neg_a=*/false, a, /*neg_b=*/false, b,
      /*c_mod=*/(short)0, c, /*reuse_a=*/false, /*reuse_b=*/false);
}

static __device__ __forceinline__ void atomAddF(float* p, float v) {
  unsafeAtomicAdd(p, v);   // hardware global_atomic_add_f32 (no CAS loop)
}

#define FDIM 128
#define ODIM 16
#define NSTEP 4  // prop_steps + 1

// ---------------------------------------------------------------------------
// 1) Per-edge complex magnetic weights.
// ---------------------------------------------------------------------------
__global__ void __launch_bounds__(256)
edge_weights_kernel(const float* __restrict__ w_sym,
                    const float* __restrict__ ent,
                    const float* __restrict__ ccf,
                    const float* __restrict__ qp,
                    const float* __restrict__ enc_w,
                    const float* __restrict__ enc_b,
                    float* __restrict__ wr, float* __restrict__ wi, int nE) {
  int e = blockIdx.x * blockDim.x + threadIdx.x;
  if (e >= nE) return;
  float theta = qp[0] * (ent[e] * enc_w[0] + ccf[e] * enc_w[1] + enc_b[0]);
  float s, c;
  sincosf(theta, &s, &c);
  float w = w_sym[e];
  wr[e] = w * c;
  wi[e] = w * s;
}

// ---------------------------------------------------------------------------
// 2) Copy input features into step-0 slices of the R/I stacks.
// ---------------------------------------------------------------------------
__global__ void __launch_bounds__(256)
init_slice0_kernel(const float* __restrict__ rf, const float* __restrict__ imf,
                   float* __restrict__ R0, float* __restrict__ I0, int n4) {
  int i = blockIdx.x * blockDim.x + threadIdx.x;
  if (i >= n4) return;
  ((float4*)R0)[i] = ((const float4*)rf)[i];
  ((float4*)I0)[i] = ((const float4*)imf)[i];
}

// ---------------------------------------------------------------------------
// 3) One complex propagation step: fused 4-SpMM.
//    One wave per edge; lane handles 4 consecutive features (float4).
// ---------------------------------------------------------------------------
__global__ void __launch_bounds__(256)
prop_step_kernel(const float* __restrict__ Rin, const float* __restrict__ Iin,
                 float* __restrict__ Rout, float* __restrict__ Iout,
                 const int* __restrict__ rows, const int* __restrict__ cols,
                 const float* __restrict__ wr, const float* __restrict__ wi,
                 int nE) {
  const int lane = threadIdx.x & 31;
  const int wid = (blockIdx.x * blockDim.x + threadIdx.x) >> 5;
  const int nw = (gridDim.x * blockDim.x) >> 5;
  for (int e = wid; e < nE; e += nw) {
    const int r = rows[e];
    const int c = cols[e];
    const float a = wr[e];
    const float b = wi[e];
    float4 rv = ((const float4*)(Rin + (size_t)c * FDIM))[lane];
    float4 iv = ((const float4*)(Iin + (size_t)c * FDIM))[lane];
    float4 orv, oiv;
    orv.x = a * rv.x - b * iv.x;  oiv.x = b * rv.x + a * iv.x;
    orv.y = a * rv.y - b * iv.y;  oiv.y = b * rv.y + a * iv.y;
    orv.z = a * rv.z - b * iv.z;  oiv.z = b * rv.z + a * iv.z;
    orv.w = a * rv.w - b * iv.w;  oiv.w = b * rv.w + a * iv.w;
    float* pr = Rout + (size_t)r * FDIM + lane * 4;
    float* pi = Iout + (size_t)r * FDIM + lane * 4;
    atomAddF(pr + 0, orv.x); atomAddF(pr + 1, orv.y);
    atomAddF(pr + 2, orv.z); atomAddF(pr + 3, orv.w);
    atomAddF(pi + 0, oiv.x); atomAddF(pi + 1, oiv.y);
    atomAddF(pi + 2, oiv.z); atomAddF(pi + 3, oiv.w);
  }
}

// ---------------------------------------------------------------------------
// 4) Attention pooling over the 4 propagation steps. One wave per node.
// ---------------------------------------------------------------------------
__global__ void __launch_bounds__(256)
attn_pool_kernel(const float* __restrict__ S,  // [4, N, F]
                 const float* __restrict__ aw, const float* __restrict__ ab,
                 float* __restrict__ out, int nNodes) {
  const int wid = (blockIdx.x * blockDim.x + threadIdx.x) >> 5;
  if (wid >= nNodes) return;
  const int lane = threadIdx.x & 31;
  const float4 w4 = ((const float4*)aw)[lane];
  const float b = ab[0];
  const size_t NF = (size_t)nNodes * FDIM;

  float4 v[NSTEP];
  float d[NSTEP];
#pragma unroll
  for (int s = 0; s < NSTEP; ++s) {
    v[s] = ((const float4*)(S + s * NF + (size_t)wid * FDIM))[lane];
    float p = v[s].x * w4.x + v[s].y * w4.y + v[s].z * w4.z + v[s].w * w4.w;
#pragma unroll
    for (int off = 16; off >= 1; off >>= 1) p += __shfl_xor(p, off, 32);
    d[s] = 1.0f / (1.0f + expf(-(p + b)));   // sigmoid
  }
  // softmax over the 4 step scores
  float m = fmaxf(fmaxf(d[0], d[1]), fmaxf(d[2], d[3]));
  float e0 = expf(d[0] - m), e1 = expf(d[1] - m);
  float e2 = expf(d[2] - m), e3 = expf(d[3] - m);
  float inv = 1.0f / (e0 + e1 + e2 + e3);
  float4 acc;
  acc.x = (e0 * v[0].x + e1 * v[1].x + e2 * v[2].x + e3 * v[3].x) * inv;
  acc.y = (e0 * v[0].y + e1 * v[1].y + e2 * v[2].y + e3 * v[3].y) * inv;
  acc.z = (e0 * v[0].z + e1 * v[1].z + e2 * v[2].z + e3 * v[3].z) * inv;
  acc.w = (e0 * v[0].w + e1 * v[1].w + e2 * v[2].w + e3 * v[3].w) * inv;
  ((float4*)(out + (size_t)wid * FDIM))[lane] = acc;
}

// ---------------------------------------------------------------------------
// 5) Fused complex MLP layer with ComReLU, fp32 WMMA (16x16x4_f32).
//    Block = 256 threads = 8 waves; block owns 16 rows, wave w owns cols
//    [16w, 16w+16). A-tiles (real+imag, 16x128 each) staged in LDS.
// ---------------------------------------------------------------------------
#define LSTR 132  // padded LDS row stride (floats) -> conflict-free frag reads

__global__ void __launch_bounds__(256)
mlp_layer_kernel(const float* __restrict__ Xr, const float* __restrict__ Xi,
                 const float* __restrict__ Wr, const float* __restrict__ br_,
                 const float* __restrict__ Wi, const float* __restrict__ bi_,
                 float* __restrict__ Yr, float* __restrict__ Yi) {
  __shared__ float Asr[16 * LSTR];
  __shared__ float Asi[16 * LSTR];
  const int t = threadIdx.x;
  const int row0 = blockIdx.x * 16;

  // Stage 16x128 real+imag tiles (512 float4 each; 2 per thread).
  for (int idx = t; idx < 16 * 32; idx += 256) {
    int rr = idx >> 5;
    int c4 = idx & 31;
    float4 vr = ((const float4*)(Xr + (size_t)(row0 + rr) * FDIM))[c4];
    float4 vi = ((const float4*)(Xi + (size_t)(row0 + rr) * FDIM))[c4];
    *(float4*)&Asr[rr * LSTR + c4 * 4] = vr;
    *(float4*)&Asi[rr * LSTR + c4 * 4] = vi;
  }
  __syncthreads();

  const int lane = t & 31;
  const int wv = t >> 5;
  const int half = lane >> 4;   // 0: K lo / rows 0-7, 1: K hi / rows 8-15
  const int mrow = lane & 15;
  const int cc = wv * 16 + mrow;  // output column

  v8f cr = {};
  v8f ci = {};
#pragma unroll 4
  for (int k = 0; k < FDIM; k += 4) {
    const int kr = k + 2 * half;
    v2f ar, ai, brf, bif;
    ar[0] = Asr[mrow * LSTR + kr];     ar[1] = Asr[mrow * LSTR + kr + 1];
    ai[0] = Asi[mrow * LSTR + kr];     ai[1] = Asi[mrow * LSTR + kr + 1];
    brf[0] = Wr[kr * FDIM + cc];       brf[1] = Wr[(kr + 1) * FDIM + cc];
    bif[0] = Wi[kr * FDIM + cc];       bif[1] = Wi[(kr + 1) * FDIM + cc];
    cr = wmma_f32_k4(ar, brf, cr);
    ci = wmma_f32_k4(ai, bif, ci);
  }

  const float bvr = br_[cc];
  const float bvi = bi_[cc];
  const int rbase = row0 + 8 * half;
#pragma unroll
  for (int j = 0; j < 8; ++j) {
    float tr = cr[j] + bvr;
    float ti = ci[j] + bvi;
    float msk = (tr >= 0.0f) ? 1.0f : 0.0f;   // ComReLU mask from real part
    Yr[(size_t)(rbase + j) * FDIM + cc] = tr * msk;
    Yi[(size_t)(rbase + j) * FDIM + cc] = ti * msk;
  }
}

// ---------------------------------------------------------------------------
// 6) Output layer: [Xr | Xi] (N,256) @ out_w (256,16) + out_b.
//    One wave per 16-row tile; K=256 split into real (rows 0..127) and
//    imag (rows 128..255) halves of out_w.
// ---------------------------------------------------------------------------
__global__ void __launch_bounds__(256)
out_layer_kernel(const float* __restrict__ Xr, const float* __restrict__ Xi,
                 const float* __restrict__ W, const float* __restrict__ b_,
                 float* __restrict__ Y, int ntiles) {
  const int wv = threadIdx.x >> 5;
  const int tile = blockIdx.x * 8 + wv;
  if (tile >= ntiles) return;
  const int lane = threadIdx.x & 31;
  const int half = lane >> 4;
  const int mrow = lane & 15;
  const int row0 = tile * 16;

  v8f c = {};
#pragma unroll 4
  for (int k = 0; k < FDIM; k += 4) {
    const int kr = k + 2 * half;
    v2f a, bb;
    const float* xp = Xr + (size_t)(row0 + mrow) * FDIM + kr;
    a[0] = xp[0]; a[1] = xp[1];
    bb[0] = W[kr * ODIM + mrow];
    bb[1] = W[(kr + 1) * ODIM + mrow];
    c = wmma_f32_k4(a, bb, c);
  }
#pragma unroll 4
  for (int k = 0; k < FDIM; k += 4) {
    const int kr = k + 2 * half;
    v2f a, bb;
    const float* xp = Xi + (size_t)(row0 + mrow) * FDIM + kr;
    a[0] = xp[0]; a[1] = xp[1];
    bb[0] = W[(FDIM + kr) * ODIM + mrow];
    bb[1] = W[(FDIM + kr + 1) * ODIM + mrow];
    c = wmma_f32_k4(a, bb, c);
  }

  const float bv = b_[mrow];
  const int rbase = row0 + 8 * half;
#pragma unroll
  for (int j = 0; j < 8; ++j) {
    Y[(size_t)(rbase + j) * ODIM + mrow] = c[j] + bv;
  }
}

// ---------------------------------------------------------------------------
// Host-side orchestration (graph-capture safe: only async ops on `stream`).
// ---------------------------------------------------------------------------
extern "C" void kernel_launch(void* const* d_in, const int* in_sizes, int n_in,
                              void* d_out, int out_size, void* d_ws,
                              size_t ws_size, hipStream_t stream) {
  const float* real_f = (const float*)d_in[0];
  const float* imag_f = (const float*)d_in[1];
  const int*   eidx   = (const int*)d_in[2];
  const float* w_sym  = (const float*)d_in[3];
  const float* ent    = (const float*)d_in[4];
  const float* ccf    = (const float*)d_in[5];
  const float* qp     = (const float*)d_in[6];
  const float* enc_w  = (const float*)d_in[7];
  const float* enc_b  = (const float*)d_in[8];
  const float* ratt_w = (const float*)d_in[9];
  const float* ratt_b = (const float*)d_in[10];
  const float* iatt_w = (const float*)d_in[11];
  const float* iatt_b = (const float*)d_in[12];
  const float* rlw0   = (const float*)d_in[13];
  const float* rlb0   = (const float*)d_in[14];
  const float* rlw1   = (const float*)d_in[15];
  const float* rlb1   = (const float*)d_in[16];
  const float* ilw0   = (const float*)d_in[17];
  const float* ilb0   = (const float*)d_in[18];
  const float* ilw1   = (const float*)d_in[19];
  const float* ilb1   = (const float*)d_in[20];
  const float* out_w  = (const float*)d_in[21];
  const float* out_b  = (const float*)d_in[22];

  const int N = in_sizes[0] / FDIM;   // 50000
  const int E = in_sizes[2] / 2;      // 800000
  const size_t NF = (size_t)N * FDIM; // 6.4M floats

  // Workspace layout (floats): ~314 MB total.
  float* ws = (float*)d_ws;
  float* wr = ws;                       // [E]
  float* wi = wr + E;                   // [E]
  float* Rs = wi + E;                   // [4, N, F]
  float* Is = Rs + NSTEP * NF;          // [4, N, F]
  float* pr = Is + NSTEP * NF;          // [N, F] pooled real / layer ping
  float* pi = pr + NF;                  // [N, F] pooled imag
  float* hr = pi + NF;                  // [N, F] hidden real
  float* hi = hr + NF;                  // [N, F] hidden imag

  const int* rows = eidx;
  const int* cols = eidx + E;

  // 1) edge weights
  edge_weights_kernel<<<(E + 255) / 256, 256, 0, stream>>>(
      w_sym, ent, ccf, qp, enc_w, enc_b, wr, wi, E);

  // 2) init step-0 slices, zero accumulation slices 1..3
  {
    int n4 = (int)(NF / 4);
    init_slice0_kernel<<<(n4 + 255) / 256, 256, 0, stream>>>(
        real_f, imag_f, Rs, Is, n4);
    hipMemsetAsync(Rs + NF, 0, (NSTEP - 1) * NF * sizeof(float), stream);
    hipMemsetAsync(Is + NF, 0, (NSTEP - 1) * NF * sizeof(float), stream);
  }

  // 3) three fused complex propagation steps
  {
    const int blocks = 50000;  // 400k waves, grid-stride over 800k edges
    for (int s = 0; s < NSTEP - 1; ++s) {
      prop_step_kernel<<<blocks, 256, 0, stream>>>(
          Rs + (size_t)s * NF, Is + (size_t)s * NF,
          Rs + (size_t)(s + 1) * NF, Is + (size_t)(s + 1) * NF,
          rows, cols, wr, wi, E);
    }
  }

  // 4) attention pooling (real and imag)
  {
    int blocks = (N + 7) / 8;  // 8 waves (nodes) per 256-thread block
    attn_pool_kernel<<<blocks, 256, 0, stream>>>(Rs, ratt_w, ratt_b, pr, N);
    attn_pool_kernel<<<blocks, 256, 0, stream>>>(Is, iatt_w, iatt_b, pi, N);
  }

  // 5) two fused complex MLP layers (WMMA f32)
  {
    int ntiles = N / 16;  // 3125 (N divisible by 16)
    mlp_layer_kernel<<<ntiles, 256, 0, stream>>>(pr, pi, rlw0, rlb0, ilw0,
                                                 ilb0, hr, hi);
    mlp_layer_kernel<<<ntiles, 256, 0, stream>>>(hr, hi, rlw1, rlb1, ilw1,
                                                 ilb1, pr, pi);
    // 6) output projection
    out_layer_kernel<<<(ntiles + 7) / 8, 256, 0, stream>>>(
        pr, pi, out_w, out_b, (float*)d_out, ntiles);
  }
}